// GNNWithEmbedding_11029476016728
// MI455X (gfx1250) — compile-verified
//
#include <hip/hip_runtime.h>
#include <hip/hip_bf16.h>

// 2-layer GCN for MI455X (gfx1250, wave32).
//
// GEMMs: V_WMMA_F32_16X16X4_F32 (full f32 precision; ~2 GFLOP total, trivial).
// Aggregation: instead of 422M scalar f32 atomics (the reference's
// scatter-add), build a dst-bucketed CSR once (count + LDS prefix scan +
// cursor fill), then gather-reduce per destination node: one wave per node,
// float4 per lane, single coalesced store. Feature matrices (25.6 MB) are
// L2-resident (192 MB L2), so gathers run at L2 bandwidth; atomic RMW traffic
// (845 MB/layer) is replaced by 25.6 MB/layer of plain stores.

typedef float v2f __attribute__((ext_vector_type(2)));
typedef float v8f __attribute__((ext_vector_type(8)));

#define NNODES 50000
#define EMB_DIM 32
#define HID 128
#define OUTD 128

// ---------------- degree count / dinv ----------------

__global__ void k_zero_i32(int* __restrict__ p, int n) {
  int i = blockIdx.x * blockDim.x + threadIdx.x;
  if (i < n) p[i] = 0;
}

__global__ void k_count(const int* __restrict__ dst, int E,
                        int* __restrict__ cnt) {
  int e = blockIdx.x * blockDim.x + threadIdx.x;
  if (e < E) {
    int d = dst[e];
    if ((unsigned)d < (unsigned)NNODES) atomicAdd(&cnt[d], 1);
  }
}

__global__ void k_dinv(const int* __restrict__ cnt, float* __restrict__ dinv,
                       int n) {
  int i = blockIdx.x * blockDim.x + threadIdx.x;
  if (i < n) dinv[i] = rsqrtf((float)(cnt[i] + 1));  // +1 self-loop
}

// ---------------- prefix scan (256/block LDS scan, 3 passes) ----------------

__global__ __launch_bounds__(256) void k_scan_block(
    const int* __restrict__ cnt, int n, int* __restrict__ incl,
    int* __restrict__ bsum) {
  __shared__ int sm[256];
  int i = blockIdx.x * 256 + threadIdx.x;
  sm[threadIdx.x] = (i < n) ? cnt[i] : 0;
  __syncthreads();
#pragma unroll
  for (int off = 1; off < 256; off <<= 1) {
    int t = (threadIdx.x >= off) ? sm[threadIdx.x - off] : 0;
    __syncthreads();
    sm[threadIdx.x] += t;
    __syncthreads();
  }
  if (i < n) incl[i] = sm[threadIdx.x];
  if (threadIdx.x == 255) bsum[blockIdx.x] = sm[255];
}

__global__ __launch_bounds__(256) void k_scan_sums(int* __restrict__ bsum,
                                                   int nb) {
  __shared__ int sm[256];
  int t = threadIdx.x;
  sm[t] = (t < nb) ? bsum[t] : 0;
  __syncthreads();
#pragma unroll
  for (int off = 1; off < 256; off <<= 1) {
    int v = (t >= off) ? sm[t - off] : 0;
    __syncthreads();
    sm[t] += v;
    __syncthreads();
  }
  if (t < nb) bsum[t] = sm[t];  // inclusive scan of block sums
}

__global__ __launch_bounds__(256) void k_scan_finalize(
    const int* __restrict__ incl, const int* __restrict__ bsumscan,
    const int* __restrict__ cnt, int n, int* __restrict__ rowptr,
    int* __restrict__ cursor) {
  int i = blockIdx.x * 256 + threadIdx.x;
  if (i >= n) return;
  int boff = (blockIdx.x == 0) ? 0 : bsumscan[blockIdx.x - 1];
  int inc = incl[i] + boff;
  rowptr[i + 1] = inc;
  cursor[i] = inc - cnt[i];  // exclusive start
  if (i == 0) rowptr[0] = 0;
}

__global__ void k_fill(const int* __restrict__ src, const int* __restrict__ dst,
                       int E, int* __restrict__ cursor, int* __restrict__ col) {
  int e = blockIdx.x * blockDim.x + threadIdx.x;
  if (e >= E) return;
  int s = src[e];
  int d = dst[e];
  if ((unsigned)s >= (unsigned)NNODES || (unsigned)d >= (unsigned)NNODES)
    return;
  int pos = atomicAdd(&cursor[d], 1);
  col[pos] = s;
}

// ---------------- WMMA GEMM: C[N x COLS] = A[N x K] @ B[K x COLS] ----------
// One wave per 16x16 tile; 8 waves/block cover a 16x128 stripe.
// Requires N % 16 == 0 (50000 = 3125*16) and COLS == 128.

template <int K, int COLS, bool GATHER>
__global__ __launch_bounds__(256) void k_gemm_wmma(
    const float* __restrict__ A, const float* __restrict__ B,
    float* __restrict__ C, const int* __restrict__ xidx) {
  const int lane = threadIdx.x & 31;
  const int wave = threadIdx.x >> 5;
  const int lo = lane & 15;  // M (A) or N (B/C) within tile
  const int hi = lane >> 4;  // K-pair / row-half select
  const int row0 = blockIdx.x * 16;
  const int col0 = wave * 16;

  int arow = row0 + lo;
  if (GATHER) arow = xidx[arow];  // emb[x] lookup
  const float* __restrict__ Ap = A + (size_t)arow * K;

  v8f acc = {};
#pragma unroll
  for (int kb = 0; kb < K; kb += 4) {
    v2f a, b;
    a.x = Ap[kb + 2 * hi];
    a.y = Ap[kb + 2 * hi + 1];
    b.x = B[(size_t)(kb + 2 * hi) * COLS + col0 + lo];
    b.y = B[(size_t)(kb + 2 * hi + 1) * COLS + col0 + lo];
    acc = __builtin_amdgcn_wmma_f32_16x16x4_f32(
        false, a, false, b, (short)0, acc, false, false);
  }
#pragma unroll
  for (int g = 0; g < 8; ++g) {
    C[(size_t)(row0 + g + 8 * hi) * COLS + col0 + lo] = acc[g];
  }
}

// ---------------- dst-centric gather-reduce (no atomics) -------------------
// One wave per destination node; lane owns 4 of 128 dims.
// out[d] = dinv[d] * ( sum_{s in in(d)} H[s]*dinv[s] + H[d]*dinv[d] ) + bias
// (optionally ReLU).

template <bool RELU>
__global__ __launch_bounds__(256) void k_aggregate(
    const int* __restrict__ rowptr, const int* __restrict__ col,
    const float* __restrict__ H, const float* __restrict__ dinv,
    const float* __restrict__ bias, float* __restrict__ out, int n) {
  int d = blockIdx.x * (blockDim.x >> 5) + (threadIdx.x >> 5);
  int lane = threadIdx.x & 31;
  if (d >= n) return;

  int k0 = rowptr[d];
  int k1 = rowptr[d + 1];
  float4 acc = make_float4(0.f, 0.f, 0.f, 0.f);
  for (int k = k0; k < k1; ++k) {
    int s = col[k];  // uniform across lanes -> single L2 request
    float w = dinv[s];
    float4 v = *((const float4*)(H + (size_t)s * 128) + lane);
    acc.x += v.x * w;
    acc.y += v.y * w;
    acc.z += v.z * w;
    acc.w += v.w * w;
  }
  float dd = dinv[d];
  float4 self = *((const float4*)(H + (size_t)d * 128) + lane);
  float4 bb = *((const float4*)bias + lane);
  float4 r;
  r.x = (acc.x + self.x * dd) * dd + bb.x;
  r.y = (acc.y + self.y * dd) * dd + bb.y;
  r.z = (acc.z + self.z * dd) * dd + bb.z;
  r.w = (acc.w + self.w * dd) * dd + bb.w;
  if (RELU) {
    r.x = fmaxf(r.x, 0.f);
    r.y = fmaxf(r.y, 0.f);
    r.z = fmaxf(r.z, 0.f);
    r.w = fmaxf(r.w, 0.f);
  }
  *((float4*)(out + (size_t)d * 128) + lane) = r;
}

// ---------------- launcher ----------------

static inline size_t align256(size_t x) { return (x + 255) & ~(size_t)255; }

extern "C" void kernel_launch(void* const* d_in, const int* in_sizes, int n_in,
                              void* d_out, int out_size, void* d_ws,
                              size_t ws_size, hipStream_t stream) {
  const int* x    = (const int*)d_in[0];
  const int* ei   = (const int*)d_in[1];
  const float* emb = (const float*)d_in[2];
  const float* W1  = (const float*)d_in[3];
  const float* b1  = (const float*)d_in[4];
  const float* W2  = (const float*)d_in[5];
  const float* b2  = (const float*)d_in[6];
  float* out = (float*)d_out;

  const int N = in_sizes[0];      // 50000 (multiple of 16)
  const int E = in_sizes[1] / 2;  // 1,600,000
  const int* src = ei;
  const int* dst = ei + E;
  const int NT = 256;
  const int nsb = (N + NT - 1) / NT;  // scan blocks (196)

  // workspace layout (all 256B-aligned)
  char* ws = (char*)d_ws;
  size_t o = 0;
  int* cnt = (int*)(ws + o);        o += align256((size_t)N * 4);
  int* incl = (int*)(ws + o);       o += align256((size_t)N * 4);
  int* bsum = (int*)(ws + o);       o += align256((size_t)nsb * 4);
  int* rowptr = (int*)(ws + o);     o += align256((size_t)(N + 1) * 4);
  int* cursor = (int*)(ws + o);     o += align256((size_t)N * 4);
  int* col = (int*)(ws + o);        o += align256((size_t)E * 4);
  float* dinv = (float*)(ws + o);   o += align256((size_t)N * 4);
  float* bufA = (float*)(ws + o);   o += align256((size_t)N * 128 * 4);
  float* bufB = (float*)(ws + o);   // H2; total ~58.8 MB

  // 1) in-degree counts, dinv
  k_zero_i32<<<nsb, NT, 0, stream>>>(cnt, N);
  k_count<<<(E + NT - 1) / NT, NT, 0, stream>>>(dst, E, cnt);
  k_dinv<<<nsb, NT, 0, stream>>>(cnt, dinv, N);

  // 2) CSR build: scan + fill
  k_scan_block<<<nsb, NT, 0, stream>>>(cnt, N, incl, bsum);
  k_scan_sums<<<1, NT, 0, stream>>>(bsum, nsb);
  k_scan_finalize<<<nsb, NT, 0, stream>>>(incl, bsum, cnt, N, rowptr, cursor);
  k_fill<<<(E + NT - 1) / NT, NT, 0, stream>>>(src, dst, E, cursor, col);

  // 3) layer 1: GEMM (K=32) then gather-reduce + bias + ReLU
  k_gemm_wmma<EMB_DIM, HID, true><<<N / 16, NT, 0, stream>>>(emb, W1, bufA, x);
  k_aggregate<true><<<(N + 7) / 8, NT, 0, stream>>>(rowptr, col, bufA, dinv, b1,
                                                    bufB, N);

  // 4) layer 2: GEMM (K=128) then gather-reduce + bias into d_out
  k_gemm_wmma<HID, OUTD, false><<<N / 16, NT, 0, stream>>>(bufB, W2, bufA,
                                                           nullptr);
  k_aggregate<false><<<(N + 7) / 8, NT, 0, stream>>>(rowptr, col, bufA, dinv,
                                                     b2, out, N);
}